// final_block_nodc_15341623181329
// MI455X (gfx1250) — compile-verified
//
#include <hip/hip_runtime.h>
#include <hip/hip_bf16.h>

// ---------------------------------------------------------------------------
// 3-level inverse 2-D db2 wavelet transform (periodized), 4096x4096 fp32.
//
// Each separable 1-D synthesis pass, per 16x16 tile:
//     W(16x16) = M(16x20) x Zg(20x16)
// M = constant banded matrix of db2 taps, Zg = 18 gathered rows (9 low-half +
// 9 high-half, circular) + 2 zero pads -> 5 accumulated V_WMMA_F32_16X16X4_F32
// (exact fp32, RNE). Result is stored transposed so the same kernel applied
// twice performs both axes with fully coalesced loads/stores.
//
// Perf model (MI455X): memory-bound, ~400 MB total @ 23.3 TB/s ~= 17 us.
// Strides are template constants: all inner-loop addressing folds into
// instruction immediates (loads offset:64*t, stores offset:t*16*DST_LD*4),
// and the updated-region select is strip-constant (strips are 128-aligned,
// U is a multiple of 128) so it hoists fully into setup. Steady state per
// tile: 10 b32 loads (claused) + 5 WMMA + 2 b128 stores, ~0 VALU.
// ---------------------------------------------------------------------------

typedef float v2f __attribute__((ext_vector_type(2)));
typedef float v4f __attribute__((ext_vector_type(4)));
typedef float v8f __attribute__((ext_vector_type(8)));

// db2 synthesis taps: LO = [d0,d1,d2,d3], HI = [d3,-d2,d1,-d0]
#define D0 0.4829629131445341f
#define D1 0.8365163037378079f
#define D2 0.2241438680420134f
#define D3 (-0.1294095225512604f)

// Constant synthesis matrix M[rho][k], rho = output row in tile (0..15),
// k = gathered-row index (0..19). q = rho>>1:
//   even rho: M[q]=LO2, M[q+1]=LO0, M[q+9]=HI2, M[q+10]=HI0
//   odd  rho: M[q]=LO3, M[q+1]=LO1, M[q+9]=HI3, M[q+10]=HI1
__device__ __forceinline__ float acoef(int q, int odd, int k) {
  float v = 0.0f;
  if (k == q)      v = odd ? D3  : D2;   // LO3 / LO2
  if (k == q + 1)  v = odd ? D1  : D0;   // LO1 / LO0
  if (k == q + 9)  v = odd ? -D0 : D1;   // HI3 / HI2
  if (k == q + 10) v = odd ? -D2 : D3;   // HI1 / HI0
  return v;
}

#define NTC 8  // column-tiles per wave (16x128 strip; 128-aligned)

// One 1-D synthesis pass along axis 0 of an SSxSS block, result stored
// transposed. Source element (r,c) comes from `upd` when r<U && c<U
// (previous level's reconstruction), else from `orig`; both use stride
// SRC_LD. U is a multiple of 128, so the column predicate is constant per
// 128-wide strip and all source pointers are fixed at setup. One wave
// computes a 16x128 strip; EXEC stays all-ones (wave-uniform control flow
// only), as WMMA requires.
template <int SRC_LD, int DST_LD, int SS>
__global__ __launch_bounds__(256) void iwt_pass(const float* __restrict__ orig,
                                                const float* __restrict__ upd,
                                                int U,
                                                float* __restrict__ dst) {
  constexpr int T = SS >> 4;               // tiles per row
  constexpr int G = T / NTC;               // strip-groups per row (pow2)
  const int lane  = threadIdx.x & 31;
  const int wave  = threadIdx.x >> 5;
  const int group = blockIdx.x * 8 + wave; // grid sized exactly: T*G/8
  const int tr = group / G;                // G compile-time -> shift
  const int cg = group - tr * G;
  const int r0    = tr << 4;
  const int cbase = cg * (16 * NTC);
  const int n = lane & 15, half = lane >> 4;
  const int q = n >> 1, odd = n & 1;       // A row handled by this lane = n
  constexpr int hs = SS >> 1;
  const int m0 = tr << 3;

  // ---- per-wave setup (invariant along the strip) ----
  // Gathered source rows: g[0..8] = low[(m0-1..m0+7) mod hs], g[9..17] = high.
  int g[20];
#pragma unroll
  for (int i = 0; i < 9; ++i) {
    int m = m0 - 1 + i + hs;
    if (m >= hs) m -= hs;                  // wraps only at m0==0 edge
    if (m >= hs) m -= hs;
    g[i]     = m;
    g[9 + i] = hs + m;
  }
  g[18] = 0; g[19] = 0;                    // K pads (coefficients are zero)

  // A fragments: lane supplies A[n][k0..k0+1] per K-chunk, k0 = 4*kc + 2*half.
  v2f a[5];
#pragma unroll
  for (int kc = 0; kc < 5; ++kc) {
    const int k0 = kc * 4 + half * 2;
    a[kc][0] = acoef(q, odd, k0);
    a[kc][1] = acoef(q, odd, k0 + 1);
  }

  // Fixed source pointers per B slot. Strip is 128-aligned and U is a
  // multiple of 128, so the column-side predicate is strip-constant.
  const bool useU = (cbase < U);
  const float* p[10];
#pragma unroll
  for (int kc = 0; kc < 5; ++kc) {
#pragma unroll
    for (int j = 0; j < 2; ++j) {
      const int k  = kc * 4 + half * 2 + j;
      const int gr = g[k];
      const float* base = (useU && gr < U) ? upd : orig;
      p[kc * 2 + j] = base + (size_t)gr * SRC_LD + cbase + n;
    }
  }

  // Transposed store base: dst row (cbase+n), cols r0+8*half.. (32B/lane).
  float* dp = dst + (size_t)(cbase + n) * DST_LD + r0 + (half << 3);

  // ---- strip loop: all addressing folds into instruction immediates ----
#pragma unroll
  for (int t = 0; t < NTC; ++t) {
    // Load phase: 10 b32 loads (clause-able, partial waits before WMMAs).
    v2f b[5];
#pragma unroll
    for (int kc = 0; kc < 5; ++kc) {
      b[kc][0] = p[kc * 2][t * 16];        // offset: 64*t bytes
      b[kc][1] = p[kc * 2 + 1][t * 16];
    }
    // Compute phase: D = A x B + C accumulation chain, fp32 exact (RNE).
    v8f acc = {};
#pragma unroll
    for (int kc = 0; kc < 5; ++kc) {
      acc = __builtin_amdgcn_wmma_f32_16x16x4_f32(
          /*neg_a=*/false, a[kc], /*neg_b=*/false, b[kc],
          /*c_mod=*/(short)0, acc, /*reuse_a=*/false, /*reuse_b=*/false);
    }
    v4f lo = { acc[0], acc[1], acc[2], acc[3] };
    v4f hi = { acc[4], acc[5], acc[6], acc[7] };
    float* dpt = dp + (size_t)t * 16 * DST_LD;  // compile-time immediate
    *reinterpret_cast<v4f*>(dpt)     = lo;
    *reinterpret_cast<v4f*>(dpt + 4) = hi;
  }
}

template <int SRC_LD, int DST_LD, int SS>
static inline void run_pass(const float* orig, const float* upd, int U,
                            float* dst, hipStream_t stream) {
  constexpr int T      = SS >> 4;
  constexpr int blocks = T * (T / NTC) / 8;  // exact: T >= 64
  iwt_pass<SRC_LD, DST_LD, SS><<<blocks, 256, 0, stream>>>(orig, upd, U, dst);
}

extern "C" void kernel_launch(void* const* d_in, const int* in_sizes, int n_in,
                              void* d_out, int out_size, void* d_ws, size_t ws_size,
                              hipStream_t stream) {
  (void)in_sizes; (void)n_in; (void)out_size; (void)ws_size;
  const float* x   = (const float*)d_in[0];   // (1,4096,4096) f32, flat
  float*       out = (float*)d_out;           // 4096*4096 f32
  float*       tmp = (float*)d_ws;            // needs S*S*4 <= 64 MB scratch

  // Levels: 1024 -> 2048 -> 4096. Pass A: wt -> t^T (into tmp); level L>1
  // reads the previous reconstruction from out's top-left UxU region, rest
  // from x. Pass B: t^T -> y (into out's top-left, row-major).
  run_pass<4096, 1024, 1024>(x,   out, 0,    tmp, stream);
  run_pass<1024, 4096, 1024>(tmp, tmp, 0,    out, stream);
  run_pass<4096, 2048, 2048>(x,   out, 1024, tmp, stream);
  run_pass<2048, 4096, 2048>(tmp, tmp, 0,    out, stream);
  run_pass<4096, 4096, 4096>(x,   out, 2048, tmp, stream);
  run_pass<4096, 4096, 4096>(tmp, tmp, 0,    out, stream);
}